// SelfEfficientGlobalPointer_10977936409143
// MI455X (gfx1250) — compile-verified
//
#include <hip/hip_runtime.h>
#include <hip/hip_bf16.h>
#include <math.h>

typedef __attribute__((ext_vector_type(2))) float v2f;
typedef __attribute__((ext_vector_type(4))) float v4f;
typedef __attribute__((ext_vector_type(4))) int   v4i;
typedef __attribute__((ext_vector_type(8))) float v8f;

#define INF_VAL 10000.0f
#define B_ 8
#define L_ 1024
#define H_ 1024
#define D_ 64     // head_size
#define NH_ 12    // heads

// ---------------------------------------------------------------------------
// Kernel 1: h = x @ W1 + b1   (M=8192, K=1024, N=128) using V_WMMA_F32_16X16X4_F32
// One wave per 16x16 output tile. A-lane = x[m0+l15, k0+2*hi+{0,1}] (float2),
// B-lane = W1[k0+2*hi+{0,1}, n0+l15].
// ---------------------------------------------------------------------------
__global__ void gemm_h_kernel(const float* __restrict__ x,
                              const float* __restrict__ W1,
                              const float* __restrict__ b1,
                              float* __restrict__ h) {
  const int lane = threadIdx.x & 31;
  const int wave = threadIdx.x >> 5;
  const int tile = blockIdx.x * 8 + wave;       // 4096 tiles total
  const int tm = tile >> 3;                     // 512 row tiles (M=8192)
  const int tn = tile & 7;                      // 8 col tiles (N=128)
  const int l15 = lane & 15;
  const int hi  = lane >> 4;

  const float* xrow = x  + (size_t)(tm * 16 + l15) * H_ + 2 * hi;
  const float* wcol = W1 + (size_t)(2 * hi) * 128 + tn * 16 + l15;

  v8f acc = {};
#pragma unroll 4
  for (int k0 = 0; k0 < H_; k0 += 4) {
    v2f a = *(const v2f*)(xrow + k0);
    v2f b;
    b.x = wcol[(size_t)k0 * 128];
    b.y = wcol[(size_t)k0 * 128 + 128];
    acc = __builtin_amdgcn_wmma_f32_16x16x4_f32(false, a, false, b,
                                                (short)0, acc, false, false);
  }
  const int col = tn * 16 + l15;
  const float bb = b1[col];
#pragma unroll
  for (int r = 0; r < 8; ++r) {
    int row = tm * 16 + r + 8 * hi;             // C/D layout: VGPR r -> M=r (+8 for hi half)
    h[(size_t)row * 128 + col] = acc[r] + bb;
  }
}

// ---------------------------------------------------------------------------
// Kernel 2: interleaved RoPE split. q_j = h_{2j}, k_j = h_{2j+1}; pair (2t,2t+1)
// rotated by angle pos * 10000^(-2t/64). One thread per (b,l,t), t in [0,32).
// ---------------------------------------------------------------------------
__global__ void rope_kernel(const float* __restrict__ h,
                            float* __restrict__ qw, float* __restrict__ kw) {
  int idx = blockIdx.x * blockDim.x + threadIdx.x;  // B*L*32 = 262144
  int t = idx & 31;
  int row = idx >> 5;                               // b*L + l
  int l = row & (L_ - 1);
  const float* hp = h + (size_t)row * 128 + 4 * t;
  float h0 = hp[0], h1 = hp[1], h2 = hp[2], h3 = hp[3];
  float inv = __powf(10000.0f, -(2.0f * (float)t) / (float)D_);
  float fr = (float)l * inv;
  float s = __sinf(fr), c = __cosf(fr);
  size_t qb = (size_t)row * D_ + 2 * t;
  qw[qb]     = h0 * c - h2 * s;
  qw[qb + 1] = h2 * c + h0 * s;
  kw[qb]     = h1 * c - h3 * s;
  kw[qb + 1] = h3 * c + h1 * s;
}

// ---------------------------------------------------------------------------
// Kernel 3: bias[b, c, l] = (h[b,l,:] @ W2[:,c] + b2[c]) * 0.5   (c-major layout)
// One thread per (b,l,c); K=128 dot. Tiny: 25 MFLOP total.
// ---------------------------------------------------------------------------
__global__ void bias_kernel(const float* __restrict__ h,
                            const float* __restrict__ W2,
                            const float* __restrict__ b2,
                            float* __restrict__ bias) {
  int idx = blockIdx.x * blockDim.x + threadIdx.x;  // 8*1024*24 = 196608
  int c = idx % 24;
  int row = idx / 24;                               // b*L + l
  int b = row >> 10, l = row & (L_ - 1);
  const float* hp = h + (size_t)row * 128;
  float sum = b2[c];
#pragma unroll 8
  for (int j = 0; j < 128; ++j) sum += hp[j] * W2[j * 24 + c];
  bias[((size_t)b * 24 + c) * L_ + l] = sum * 0.5f;
}

// ---------------------------------------------------------------------------
// Kernel 4: S = (qw @ kw^T)/8 via WMMA f32 16x16x4 (K=64). The S tile is
// transposed through LDS (same-wave exchange, DS in-order) so each lane owns
// two rows x 4 consecutive n -> b128 nontemporal stores. Mask (B,1,L,L) is
// written once per tile; the 12-head loop only adds biases and streams out.
// ---------------------------------------------------------------------------
__global__ void logits_kernel(const float* __restrict__ qw,
                              const float* __restrict__ kw,
                              const float* __restrict__ bias,
                              const int* __restrict__ am,
                              float* __restrict__ out) {
  __shared__ float lds[8 * 256];
  const int lane = threadIdx.x & 31;
  const int wave = threadIdx.x >> 5;
  const int b = blockIdx.y;
  const int tile = blockIdx.x * 8 + wave;       // 4096 tiles per batch (64x64)
  const int tm = tile >> 6, tn = tile & 63;
  const int l15 = lane & 15, hi = lane >> 4;
  const int m0 = tm * 16, n0 = tn * 16;

  const float* qp = qw + (size_t)(b * L_ + m0 + l15) * D_ + 2 * hi;
  const float* kp = kw + (size_t)(b * L_ + n0 + l15) * D_ + 2 * hi;

  v8f acc = {};
#pragma unroll
  for (int k0 = 0; k0 < D_; k0 += 4) {
    v2f a  = *(const v2f*)(qp + k0);
    v2f bb = *(const v2f*)(kp + k0);
    acc = __builtin_amdgcn_wmma_f32_16x16x4_f32(false, a, false, bb,
                                                (short)0, acc, false, false);
  }

  // ---- LDS transpose of the 16x16 tile (scaled) -------------------------
  float* tp = lds + wave * 256;
#pragma unroll
  for (int r = 0; r < 8; ++r)
    tp[(r + 8 * hi) * 16 + l15] = acc[r] * 0.125f;   // scale = 1/sqrt(64)
  asm volatile("" ::: "memory");  // keep DS write->read order; HW DS is in-order per wave

  const int mA = lane >> 2;            // 0..7
  const int nl = (lane & 3) * 4;       // 0,4,8,12
  v4f s0 = *(const v4f*)(tp + mA * 16 + nl);         // row m0+mA
  v4f s1 = *(const v4f*)(tp + (mA + 8) * 16 + nl);   // row m0+mA+8

  const int m_lo = m0 + mA, m_up = m0 + mA + 8;
  const int nb = n0 + nl;
  v4i amn = *(const v4i*)(am + (size_t)b * L_ + nb);
  const int am0 = am[(size_t)b * L_ + m_lo];
  const int am1 = am[(size_t)b * L_ + m_up];

  v4f pen0, pen1, msk0, msk1;
#pragma unroll
  for (int j = 0; j < 4; ++j) {
    int n = nb + j;
    float p0 = (float)(1 - am0 * amn[j]) + ((m_lo > n) ? 1.0f : 0.0f);
    float p1 = (float)(1 - am1 * amn[j]) + ((m_up > n) ? 1.0f : 0.0f);
    pen0[j] = p0 * INF_VAL;
    pen1[j] = p1 * INF_VAL;
    msk0[j] = (p0 > 0.0f) ? 1.0f : 0.0f;
    msk1[j] = (p1 > 0.0f) ? 1.0f : 0.0f;
  }

  // ---- mask output: shape (B, 1, L, L), written once per tile -----------
  const size_t maskBase = (size_t)B_ * NH_ * L_ * L_;
  __builtin_nontemporal_store(msk0, (v4f*)(out + maskBase + ((size_t)b * L_ + m_lo) * L_ + nb));
  __builtin_nontemporal_store(msk1, (v4f*)(out + maskBase + ((size_t)b * L_ + m_up) * L_ + nb));

  // ---- 12-head fan-out: logits[b,hh,m,n] --------------------------------
#pragma unroll 1
  for (int hh = 0; hh < NH_; ++hh) {
    v4f bA = *(const v4f*)(bias + ((size_t)b * 24 + 2 * hh) * L_ + nb);
    const float* b1p = bias + ((size_t)b * 24 + 2 * hh + 1) * L_;
    const float bB0 = b1p[m_lo], bB1 = b1p[m_up];
    v4f v0 = s0 + bA + bB0 - pen0;
    v4f v1 = s1 + bA + bB1 - pen1;
    const size_t ob = (size_t)(b * NH_ + hh) * L_ * L_;
    __builtin_nontemporal_store(v0, (v4f*)(out + ob + (size_t)m_lo * L_ + nb));
    __builtin_nontemporal_store(v1, (v4f*)(out + ob + (size_t)m_up * L_ + nb));
  }
}

// ---------------------------------------------------------------------------
extern "C" void kernel_launch(void* const* d_in, const int* in_sizes, int n_in,
                              void* d_out, int out_size, void* d_ws, size_t ws_size,
                              hipStream_t stream) {
  const float* x   = (const float*)d_in[0];
  const int*   am  = (const int*)d_in[1];
  const float* W1  = (const float*)d_in[2];
  const float* b1  = (const float*)d_in[3];
  const float* W2  = (const float*)d_in[4];
  const float* b2  = (const float*)d_in[5];
  float* out = (float*)d_out;

  // workspace layout (floats): h (1M), qw (512K), kw (512K), bias (192K) ~ 8.75 MB
  float* h    = (float*)d_ws;
  float* qwp  = h   + (size_t)B_ * L_ * 128;
  float* kwp  = qwp + (size_t)B_ * L_ * D_;
  float* bsp  = kwp + (size_t)B_ * L_ * D_;

  // 1) h = x @ W1 + b1 : 4096 tiles, 8 waves/block
  gemm_h_kernel<<<512, 256, 0, stream>>>(x, W1, b1, h);
  // 2) RoPE split: 262144 threads
  rope_kernel<<<1024, 256, 0, stream>>>(h, qwp, kwp);
  // 3) bias GEMM: 196608 threads
  bias_kernel<<<768, 256, 0, stream>>>(h, W2, b2, bsp);
  // 4) logits + mask: per batch 4096 tiles, 8 waves/block
  logits_kernel<<<dim3(512, B_), 256, 0, stream>>>(qwp, kwp, bsp, am, out);
}